// TAA_63127429317243
// MI455X (gfx1250) — compile-verified
//
#include <hip/hip_runtime.h>
#include <hip/hip_bf16.h>

// TAA forward for MI455X (gfx1250).
// Memory-bound (~365 MB min traffic -> ~16us floor at 23.3 TB/s).
// CDNA5-specific path: Tensor Data Mover (tensor_load_to_lds) stages an
// 18x18x3 halo tile of x into LDS (one DMA per block, issued by wave 0,
// synchronized with s_wait_tensorcnt + workgroup barrier).
// History gathers reduced to 12/channel by exploiting the integer-coordinate
// taps of the 5-tap Catmull-Rom plus pattern. mv/out use NT cache hints so
// L2 stays resident with the history planes the gathers depend on.

typedef unsigned int u32x4 __attribute__((ext_vector_type(4)));
typedef int          i32x8 __attribute__((ext_vector_type(8)));
typedef int          i32x4 __attribute__((ext_vector_type(4)));
typedef float        f32x2 __attribute__((ext_vector_type(2)));

#define TILE   16
#define HALO   18           // TILE + 2 halo
#define CH_STR (HALO*HALO)  // 324 floats per channel in LDS

__global__ __launch_bounds__(256)
void taa_fwd_kernel(const float* __restrict__ x,
                    const float* __restrict__ mv,
                    const float* __restrict__ history,
                    float* __restrict__ out,
                    int B, int C, int H, int W) {
  const int plane = H * W;
  const int b  = blockIdx.z;
  const int w0 = blockIdx.x * TILE;
  const int h0 = blockIdx.y * TILE;
  const int tx = threadIdx.x, ty = threadIdx.y;
  const int wpx = w0 + tx, hpx = h0 + ty;

  __shared__ float tile[3 * CH_STR];

  // Halo tile origin, clamped so the 18x18 window is fully in-bounds.
  // Clamped-neighbor indexing == SAME-padded min/max pooling (idempotent).
  const int tX0 = min(max(w0 - 1, 0), W - HALO);
  const int tY0 = min(max(h0 - 1, 0), H - HALO);

  const float* xb = x + (size_t)b * C * plane;

  // --- TDM: DMA the 18x18x3 x-tile into LDS; issue from wave 0 only. ---
  const int flat = ty * TILE + tx;
  if (__builtin_amdgcn_readfirstlane(flat) == 0) {
    unsigned long long ga =
        (unsigned long long)(uintptr_t)xb +
        4ull * ((unsigned long long)tY0 * (unsigned)W + (unsigned)tX0);
    unsigned int ldsa = (unsigned int)(uintptr_t)(&tile[0]); // low 32b of flat = LDS offset

    u32x4 g0;
    g0[0] = 1u;                                   // count = 1 (valid descriptor)
    g0[1] = ldsa;                                 // lds_addr
    g0[2] = (unsigned int)(ga & 0xFFFFFFFFull);   // global_addr[31:0]
    g0[3] = ((unsigned int)(ga >> 32) & 0x01FFFFFFu) | (2u << 30); // addr[56:32], type=2

    i32x8 g1;
    g1[0] = (int)(2u << 16);                              // data_size = 4B; no flags
    g1[1] = (int)(((unsigned)W & 0xFFFFu) << 16);         // tensor_dim0[15:0]
    g1[2] = (int)((((unsigned)W >> 16) & 0xFFFFu) |
                  (((unsigned)H & 0xFFFFu) << 16));       // dim0 hi | dim1 lo
    g1[3] = (int)((((unsigned)H >> 16) & 0xFFFFu) |
                  ((unsigned)HALO << 16));                // dim1 hi | tile_dim0=18
    g1[4] = (int)((unsigned)HALO | (3u << 16));           // tile_dim1=18 | tile_dim2=3
    g1[5] = W;                                            // tensor_dim0_stride lo32
    g1[6] = (int)(((unsigned)plane & 0xFFFFu) << 16);     // dim0_str hi=0 | dim1_str lo16
    g1[7] = (int)((unsigned)plane >> 16);                 // dim1_str[47:16]

    i32x4 g2;
    g2[0] = C;   // tensor_dim2 = 3 channels
    g2[1] = 0;   // tensor_dim3 (unused)
    g2[2] = 0;   // tensor_dim2_stride (unused, tile_dim3=0)
    g2[3] = 0;   // tile_dim3 = 0

    i32x4 g3;
    g3[0] = 0; g3[1] = 0; g3[2] = 0; g3[3] = 0;

#if __has_include(<hip/amd_detail/amd_gfx1250_TDM.h>)
    // amdgpu-toolchain (clang-23): 6-arg form with extra group.
    i32x8 gx; gx[0]=0; gx[1]=0; gx[2]=0; gx[3]=0; gx[4]=0; gx[5]=0; gx[6]=0; gx[7]=0;
    __builtin_amdgcn_tensor_load_to_lds(g0, g1, g2, g3, gx, 0);
#else
    // ROCm 7.2 (clang-22): 5-arg form.
    __builtin_amdgcn_tensor_load_to_lds(g0, g1, g2, g3, 0);
#endif
    __builtin_amdgcn_s_wait_tensorcnt(0);
  }
  __syncthreads();

  if (hpx < H) {
    // --- motion vector (coalesced b64 NT load; read-once stream) ---
    const f32x2* mp =
        (const f32x2*)(mv + 2 * ((size_t)b * plane + (size_t)hpx * W + wpx));
    const f32x2 g = __builtin_nontemporal_load(mp);

    // --- Catmull-Rom weights per axis (matches reference math) ---
    const float rW = 1.0f / (float)W, rH = 1.0f / (float)H;
    float ux = (g.x + 1.0f) * 0.5f, uy = (g.y + 1.0f) * 0.5f;
    float posx = ux * (float)W,     posy = uy * (float)H;
    float px0f = floorf(posx - 0.5f), py0f = floorf(posy - 0.5f);
    float cposx = px0f + 0.5f, cposy = py0f + 0.5f;
    float fx = posx - cposx, fy = posy - cposy;
    float fx2 = fx * fx, fx3 = fx2 * fx;
    float fy2 = fy * fy, fy3 = fy2 * fy;
    float w0x = -0.5f * fx3 + fx2 - 0.5f * fx;
    float w1x =  1.5f * fx3 - 2.5f * fx2 + 1.0f;
    float w2x = -1.5f * fx3 + 2.0f * fx2 + 0.5f * fx;
    float w3x =  0.5f * fx3 - 0.5f * fx2;
    float w0y = -0.5f * fy3 + fy2 - 0.5f * fy;
    float w1y =  1.5f * fy3 - 2.5f * fy2 + 1.0f;
    float w2y = -1.5f * fy3 + 2.0f * fy2 + 0.5f * fy;
    float w3y =  0.5f * fy3 - 0.5f * fy2;
    float w12x = w1x + w2x, w12y = w1y + w2y;
    float tc12x = (cposx + w2x / w12x) * rW;
    float tc12y = (cposy + w2y / w12y) * rH;

    float k1 = w12x * w0y;   // tap (tc12x, row py0-1)
    float k2 = w0x  * w12y;  // tap (col px0-1, tc12y)
    float k3 = w3x  * w12y;  // tap (col px0+2, tc12y)
    float k4 = w12x * w3y;   // tap (tc12x, row py0+2)
    float k5 = w12x * w12y;  // tap (tc12x, tc12y)

    // Fractional x pair (shared by taps k1,k4,k5), zero-pad folded into wts.
    float ixc = tc12x * (float)W - 0.5f;
    float x0f = floorf(ixc);
    float wx1 = ixc - x0f, wx0 = 1.0f - wx1;
    int   x0  = (int)x0f, x1 = x0 + 1;
    wx0 *= (x0 >= 0 && x0 < W) ? 1.0f : 0.0f;
    wx1 *= (x1 >= 0 && x1 < W) ? 1.0f : 0.0f;
    const int cx0 = min(max(x0, 0), W - 1), cx1 = min(max(x1, 0), W - 1);

    // Fractional y pair (shared by taps k2,k3,k5).
    float iyc = tc12y * (float)H - 0.5f;
    float y0f = floorf(iyc);
    float wy1 = iyc - y0f, wy0 = 1.0f - wy1;
    int   y0  = (int)y0f, y1 = y0 + 1;
    wy0 *= (y0 >= 0 && y0 < H) ? 1.0f : 0.0f;
    wy1 *= (y1 >= 0 && y1 < H) ? 1.0f : 0.0f;
    const int icy0 = min(max(y0, 0), H - 1) * W;
    const int icy1 = min(max(y1, 0), H - 1) * W;

    // Integer-coordinate rows/cols of the plus pattern.
    const int px0 = (int)px0f, py0 = (int)py0f;
    const int rowT = py0 - 1, rowB = py0 + 2;
    const int colL = px0 - 1, colR = px0 + 2;
    const float K1 = k1 * ((rowT >= 0 && rowT < H) ? 1.0f : 0.0f);
    const float K4 = k4 * ((rowB >= 0 && rowB < H) ? 1.0f : 0.0f);
    const float K2 = k2 * ((colL >= 0 && colL < W) ? 1.0f : 0.0f);
    const float K3 = k3 * ((colR >= 0 && colR < W) ? 1.0f : 0.0f);
    const int irt = min(max(rowT, 0), H - 1) * W;
    const int irb = min(max(rowB, 0), H - 1) * W;
    const int ccl = min(max(colL, 0), W - 1);
    const int ccr = min(max(colR, 0), W - 1);

    // --- 12 gathers per channel (L2-resident) ---
    const float* hb = history + (size_t)b * C * plane;
    const float rdenom = 1.0f / (k1 + k2 + k3 + k4 + k5);
    float rep3[3];
#pragma unroll
    for (int c = 0; c < 3; ++c) {
      const float* p = hb + (size_t)c * plane;
      float t1 = wx0 * p[irt  + cx0] + wx1 * p[irt  + cx1];           // top row
      float t4 = wx0 * p[irb  + cx0] + wx1 * p[irb  + cx1];           // bottom row
      float t2 = wy0 * p[icy0 + ccl] + wy1 * p[icy1 + ccl];           // left col
      float t3 = wy0 * p[icy0 + ccr] + wy1 * p[icy1 + ccr];           // right col
      float t5 = wy0 * (wx0 * p[icy0 + cx0] + wx1 * p[icy0 + cx1]) +  // center 2x2
                 wy1 * (wx0 * p[icy1 + cx0] + wx1 * p[icy1 + cx1]);
      float rep = (K1 * t1 + K2 * t2 + K3 * t3 + K4 * t4 + k5 * t5) * rdenom;
      rep3[c] = fminf(fmaxf(rep, 0.0f), 1.0f);
    }

    // --- neighborhood clamp (from TDM-staged LDS tile) + blend ---
    const int lyc = hpx - tY0, lxc = wpx - tX0;
#pragma unroll
    for (int c = 0; c < 3; ++c) {
      const float* tc = &tile[c * CH_STR];
      float mn =  1e30f, mx = -1e30f;
#pragma unroll
      for (int dy = -1; dy <= 1; ++dy) {
        const int gy = min(max(hpx + dy, 0), H - 1) - tY0;
#pragma unroll
        for (int dx = -1; dx <= 1; ++dx) {
          const int gx2 = min(max(wpx + dx, 0), W - 1) - tX0;
          const float vv = tc[gy * HALO + gx2];
          mn = fminf(mn, vv);
          mx = fmaxf(mx, vv);
        }
      }
      const float rep = fminf(fmaxf(rep3[c], mn), mx);
      const float xv  = tc[lyc * HALO + lxc];
      // Write-once output: NT store keeps L2 for history gathers.
      __builtin_nontemporal_store(
          0.1f * xv + 0.9f * rep,
          &out[((size_t)b * C + c) * plane + (size_t)hpx * W + wpx]);
    }
  }
}

extern "C" void kernel_launch(void* const* d_in, const int* in_sizes, int n_in,
                              void* d_out, int out_size, void* d_ws, size_t ws_size,
                              hipStream_t stream) {
  const float* x       = (const float*)d_in[0];  // [B,C,H,W]
  const float* mv      = (const float*)d_in[1];  // [B,H,W,2]
  const float* history = (const float*)d_in[2];  // [B,C,H,W]
  float* out = (float*)d_out;                    // [B,C,H,W]

  const int B = 4, C = 3, H = 1080, W = 1920;
  dim3 block(TILE, TILE, 1);
  dim3 grid(W / TILE, (H + TILE - 1) / TILE, B);
  taa_fwd_kernel<<<grid, block, 0, stream>>>(x, mv, history, out, B, C, H, W);
}